// fourier_attention_step_process_layer_28114855920437
// MI455X (gfx1250) — compile-verified
//
#include <hip/hip_runtime.h>
#include <hip/hip_bf16.h>
#include <stdint.h>

// ---------------------------------------------------------------------------
// fourier_attention collapses analytically:
//   fftn(fftn(x))[i]  = S * x[rev(i)],  ifftn(ifftn(y))[i] = (1/S) * y[rev(i)]
// with S = 128^3 = 2^21 and rev(rev(i)) = i, so the whole pipeline is
// elementwise in keys[i] plus per-sample reductions (sum TL/TU, softmax).
// Bandwidth-bound. L2 strategy (192 MB): keys+query (134 MB) are read by
// passes 2/3/4 -> keep them L2-resident with default (RT) caching; TU/TL and
// values/out are single-touch -> non-temporal so they don't evict keys/query.
// Pass 1 uses async global->LDS double buffering + v_wmma_f32_16x16x4_f32
// (B = ones) as a 64-floats-per-instruction summation engine.
// ---------------------------------------------------------------------------

typedef __attribute__((ext_vector_type(2))) float v2f;
typedef __attribute__((ext_vector_type(4))) float f4;
typedef __attribute__((ext_vector_type(8))) float v8f;

#define NPER        2097152u          // 128^3 elements per sample
#define NPER4       524288u           // in float4
#define SCALE       2097152.0f
#define INV_SCALE   (1.0f / 2097152.0f)

// float offsets inside workspace
#define WS_FEAT_TL  0                 // [8]
#define WS_FEAT_TU  8                 // [8]
#define WS_MAX      16                // [8]
#define WS_INVSUM   24                // [8]
#define WS_PTL      32                // [8*64]
#define WS_PTU      (32 + 512)        // [8*64]
#define WS_PMAX     (32 + 1024)       // [8*256]
#define WS_PSUM     (32 + 1024 + 2048)// [8*256]

#if defined(__has_builtin)
#if __has_builtin(__builtin_amdgcn_wmma_f32_16x16x4_f32)
#define HAVE_WMMA4 1
#endif
#endif

// one per-lane 16B async copy: global -> LDS (ASYNCcnt), non-temporal:
// TU/TL are single-touch streams, keep them out of L2 so keys/query can stay.
__device__ __forceinline__ void async_b128_to_lds_nt(const float* g, const void* lds) {
  unsigned loff = (unsigned)(uintptr_t)lds;   // low 32 bits of generic addr == LDS byte offset
  asm volatile("global_load_async_to_lds_b128 %0, %1, off th:TH_LOAD_NT"
               :: "v"(loff), "v"(g) : "memory");
}

__device__ __forceinline__ float qk_elem(float k, float q, float TLm, float TUm,
                                         float aL, float aU, float bL, float bU) {
  float r = SCALE * k;
  float A = r * fmaxf(TUm - r, aU) / (bU * TUm - r);
  float B = r * fmaxf(r - TLm, aL) / (r - bL * TLm);
  float g = A + B - r;                 // r - tmp_max - tmp_min
  return fabsf(g) * INV_SCALE + q;     // |iFT| + query
}

// ---------------- Pass 1: per-sample sums of TU and TL ----------------------
// grid (64, 8), 256 threads. Each wave streams 4096 floats per tensor through
// a private double-buffered LDS stage (async copies), summing with WMMA.
__global__ __launch_bounds__(256) void k_pass1_sums(const float* __restrict__ TU,
                                                    const float* __restrict__ TL,
                                                    float* __restrict__ ws) {
  __shared__ f4 stage[2][2][8][32];        // [tensor][buf][wave][lane] = 16 KB
  __shared__ float wsum[2][8];

  const int tid  = threadIdx.x;
  const int lane = tid & 31;
  const int wv   = tid >> 5;
  const int samp = blockIdx.y;
  const int blk  = blockIdx.x;             // 0..63

  const unsigned base = samp * NPER + (unsigned)blk * 32768u + (unsigned)wv * 4096u;
  const float* srcs[2];
  srcs[0] = TU + base;
  srcs[1] = TL + base;

#ifdef HAVE_WMMA4
  v8f acc[2] = {};
  v2f ones; ones[0] = 1.0f; ones[1] = 1.0f;
#else
  float fsum[2] = {0.0f, 0.0f};
#endif

  // prologue: fill buffer 0 (ASYNCcnt += 2)
  async_b128_to_lds_nt(srcs[0] + lane * 4u, &stage[0][0][wv][lane]);
  async_b128_to_lds_nt(srcs[1] + lane * 4u, &stage[1][0][wv][lane]);

  for (int s = 0; s < 32; ++s) {
    const int cur = s & 1;
    if (s + 1 < 32) {
      const int nxt = cur ^ 1;
      async_b128_to_lds_nt(srcs[0] + (unsigned)(s + 1) * 128u + lane * 4u, &stage[0][nxt][wv][lane]);
      async_b128_to_lds_nt(srcs[1] + (unsigned)(s + 1) * 128u + lane * 4u, &stage[1][nxt][wv][lane]);
      asm volatile("s_wait_asynccnt 0x2" ::: "memory");  // current buffer landed
    } else {
      asm volatile("s_wait_asynccnt 0x0" ::: "memory");
    }
#pragma unroll
    for (int t = 0; t < 2; ++t) {
      f4 d = stage[t][cur][wv][lane];
#ifdef HAVE_WMMA4
      v2f a0; a0[0] = d.x; a0[1] = d.y;
      v2f a1; a1[0] = d.z; a1[1] = d.w;
      // D = A(16x4) * ones(4x16) + C  ->  every column of D holds row-sums
      acc[t] = __builtin_amdgcn_wmma_f32_16x16x4_f32(false, a0, false, ones,
                                                     (short)0, acc[t], false, false);
      acc[t] = __builtin_amdgcn_wmma_f32_16x16x4_f32(false, a1, false, ones,
                                                     (short)0, acc[t], false, false);
#else
      fsum[t] += d.x + d.y + d.z + d.w;
#endif
    }
  }

#pragma unroll
  for (int t = 0; t < 2; ++t) {
#ifdef HAVE_WMMA4
    // lane j<16 VGPR v holds D[v][j], lanes>=16 hold D[v+8][j-16]
    float sl = acc[t][0] + acc[t][1] + acc[t][2] + acc[t][3] +
               acc[t][4] + acc[t][5] + acc[t][6] + acc[t][7];
    float wtot = __shfl(sl, 0) + __shfl(sl, 16);   // rows 0..7 + rows 8..15
#else
    float v = fsum[t];
    for (int o = 16; o; o >>= 1) v += __shfl_xor(v, o);
    float wtot = v;
#endif
    if (lane == 0) wsum[t][wv] = wtot;
  }
  __syncthreads();
  if (tid == 0) {
    float t0 = 0.0f, t1 = 0.0f;
    for (int w = 0; w < 8; ++w) { t0 += wsum[0][w]; t1 += wsum[1][w]; }
    ws[WS_PTU + samp * 64 + blk] = t0;
    ws[WS_PTL + samp * 64 + blk] = t1;
  }
}

// reduce 64 block partials -> feat_TL/feat_TU per sample (8 blocks x 64 thr)
__global__ __launch_bounds__(64) void k_pass1b(float* __restrict__ ws) {
  __shared__ float sh[64];
  const int samp = blockIdx.x, t = threadIdx.x;
  sh[t] = ws[WS_PTL + samp * 64 + t];
  __syncthreads();
  for (int o = 32; o; o >>= 1) { if (t < o) sh[t] += sh[t + o]; __syncthreads(); }
  if (t == 0) ws[WS_FEAT_TL + samp] = sh[0];
  __syncthreads();
  sh[t] = ws[WS_PTU + samp * 64 + t];
  __syncthreads();
  for (int o = 32; o; o >>= 1) { if (t < o) sh[t] += sh[t + o]; __syncthreads(); }
  if (t == 0) ws[WS_FEAT_TU + samp] = sh[0];
}

// ---------------- Pass 2: per-sample max of qk ------------------------------
// keys/query loaded with default (RT) policy: 134 MB working set fits the
// 192 MB L2 and is re-read by passes 3 and 4.
__global__ __launch_bounds__(256) void k_pass2_max(const f4* __restrict__ keys,
                                                   const f4* __restrict__ query,
                                                   const float* __restrict__ aLp,
                                                   const float* __restrict__ aUp,
                                                   const float* __restrict__ bLp,
                                                   const float* __restrict__ bUp,
                                                   float* __restrict__ ws) {
  const int samp = blockIdx.y, blk = blockIdx.x, tid = threadIdx.x;
  const float TLm = ws[WS_FEAT_TL + samp], TUm = ws[WS_FEAT_TU + samp];
  const float aL = *aLp, aU = *aUp, bL = *bLp, bU = *bUp;
  const unsigned base4 = samp * NPER4 + (unsigned)blk * 2048u;

  float mx = -INFINITY;
#pragma unroll
  for (int i = 0; i < 8; ++i) {
    const unsigned idx = base4 + (unsigned)i * 256u + tid;
    f4 k = keys[idx], q = query[idx];
    mx = fmaxf(mx, qk_elem(k.x, q.x, TLm, TUm, aL, aU, bL, bU));
    mx = fmaxf(mx, qk_elem(k.y, q.y, TLm, TUm, aL, aU, bL, bU));
    mx = fmaxf(mx, qk_elem(k.z, q.z, TLm, TUm, aL, aU, bL, bU));
    mx = fmaxf(mx, qk_elem(k.w, q.w, TLm, TUm, aL, aU, bL, bU));
  }
  for (int o = 16; o; o >>= 1) mx = fmaxf(mx, __shfl_xor(mx, o));
  __shared__ float sm[8];
  if ((tid & 31) == 0) sm[tid >> 5] = mx;
  __syncthreads();
  if (tid == 0) {
    float m = sm[0];
    for (int w = 1; w < 8; ++w) m = fmaxf(m, sm[w]);
    ws[WS_PMAX + samp * 256 + blk] = m;
  }
}

__global__ __launch_bounds__(256) void k_pass2b(float* __restrict__ ws) {
  __shared__ float sh[256];
  const int samp = blockIdx.x, t = threadIdx.x;
  sh[t] = ws[WS_PMAX + samp * 256 + t];
  __syncthreads();
  for (int o = 128; o; o >>= 1) { if (t < o) sh[t] = fmaxf(sh[t], sh[t + o]); __syncthreads(); }
  if (t == 0) ws[WS_MAX + samp] = sh[0];
}

// ---------------- Pass 3: per-sample sum of exp(qk - max) -------------------
__global__ __launch_bounds__(256) void k_pass3_sum(const f4* __restrict__ keys,
                                                   const f4* __restrict__ query,
                                                   const float* __restrict__ aLp,
                                                   const float* __restrict__ aUp,
                                                   const float* __restrict__ bLp,
                                                   const float* __restrict__ bUp,
                                                   float* __restrict__ ws) {
  const int samp = blockIdx.y, blk = blockIdx.x, tid = threadIdx.x;
  const float TLm = ws[WS_FEAT_TL + samp], TUm = ws[WS_FEAT_TU + samp];
  const float mxm = ws[WS_MAX + samp];
  const float aL = *aLp, aU = *aUp, bL = *bLp, bU = *bUp;
  const unsigned base4 = samp * NPER4 + (unsigned)blk * 2048u;

  float s = 0.0f;
#pragma unroll
  for (int i = 0; i < 8; ++i) {
    const unsigned idx = base4 + (unsigned)i * 256u + tid;
    f4 k = keys[idx], q = query[idx];
    s += expf(qk_elem(k.x, q.x, TLm, TUm, aL, aU, bL, bU) - mxm);
    s += expf(qk_elem(k.y, q.y, TLm, TUm, aL, aU, bL, bU) - mxm);
    s += expf(qk_elem(k.z, q.z, TLm, TUm, aL, aU, bL, bU) - mxm);
    s += expf(qk_elem(k.w, q.w, TLm, TUm, aL, aU, bL, bU) - mxm);
  }
  for (int o = 16; o; o >>= 1) s += __shfl_xor(s, o);
  __shared__ float sm[8];
  if ((tid & 31) == 0) sm[tid >> 5] = s;
  __syncthreads();
  if (tid == 0) {
    float t0 = 0.0f;
    for (int w = 0; w < 8; ++w) t0 += sm[w];
    ws[WS_PSUM + samp * 256 + blk] = t0;
  }
}

__global__ __launch_bounds__(256) void k_pass3b(float* __restrict__ ws) {
  __shared__ float sh[256];
  const int samp = blockIdx.x, t = threadIdx.x;
  sh[t] = ws[WS_PSUM + samp * 256 + t];
  __syncthreads();
  for (int o = 128; o; o >>= 1) { if (t < o) sh[t] += sh[t + o]; __syncthreads(); }
  if (t == 0) ws[WS_INVSUM + samp] = 1.0f / sh[0];
}

// ---------------- Pass 4: out = values * softmax ----------------------------
// values (single-touch) and out (write-once) are non-temporal so they don't
// evict the L2-resident keys/query.
__global__ __launch_bounds__(256) void k_pass4_out(const f4* __restrict__ keys,
                                                   const f4* __restrict__ query,
                                                   const f4* __restrict__ values,
                                                   const float* __restrict__ aLp,
                                                   const float* __restrict__ aUp,
                                                   const float* __restrict__ bLp,
                                                   const float* __restrict__ bUp,
                                                   const float* __restrict__ ws,
                                                   f4* __restrict__ out) {
  const int samp = blockIdx.y, blk = blockIdx.x, tid = threadIdx.x;
  const float TLm = ws[WS_FEAT_TL + samp], TUm = ws[WS_FEAT_TU + samp];
  const float mxm = ws[WS_MAX + samp], inv = ws[WS_INVSUM + samp];
  const float aL = *aLp, aU = *aUp, bL = *bLp, bU = *bUp;
  const unsigned base4 = samp * NPER4 + (unsigned)blk * 2048u;

#pragma unroll
  for (int i = 0; i < 8; ++i) {
    const unsigned idx = base4 + (unsigned)i * 256u + tid;
    f4 k = keys[idx], q = query[idx];
    f4 v = __builtin_nontemporal_load(&values[idx]);
    f4 o;
    o.x = v.x * expf(qk_elem(k.x, q.x, TLm, TUm, aL, aU, bL, bU) - mxm) * inv;
    o.y = v.y * expf(qk_elem(k.y, q.y, TLm, TUm, aL, aU, bL, bU) - mxm) * inv;
    o.z = v.z * expf(qk_elem(k.z, q.z, TLm, TUm, aL, aU, bL, bU) - mxm) * inv;
    o.w = v.w * expf(qk_elem(k.w, q.w, TLm, TUm, aL, aU, bL, bU) - mxm) * inv;
    __builtin_nontemporal_store(o, &out[idx]);
  }
}

// ---------------------------------------------------------------------------
extern "C" void kernel_launch(void* const* d_in, const int* in_sizes, int n_in,
                              void* d_out, int out_size, void* d_ws, size_t ws_size,
                              hipStream_t stream) {
  const float*  query  = (const float*)d_in[0];
  const float*  keys   = (const float*)d_in[1];
  const float*  TU     = (const float*)d_in[2];
  const float*  TL     = (const float*)d_in[3];
  const float*  values = (const float*)d_in[4];
  const float*  aL     = (const float*)d_in[5];
  const float*  aU     = (const float*)d_in[6];
  const float*  bL     = (const float*)d_in[7];
  const float*  bU     = (const float*)d_in[8];
  float*        ws     = (float*)d_ws;
  float*        out    = (float*)d_out;

  const dim3 g1(64, 8), g2(256, 8);
  k_pass1_sums<<<g1, 256, 0, stream>>>(TU, TL, ws);
  k_pass1b   <<<8, 64, 0, stream>>>(ws);
  k_pass2_max<<<g2, 256, 0, stream>>>((const f4*)keys, (const f4*)query,
                                      aL, aU, bL, bU, ws);
  k_pass2b   <<<8, 256, 0, stream>>>(ws);
  k_pass3_sum<<<g2, 256, 0, stream>>>((const f4*)keys, (const f4*)query,
                                      aL, aU, bL, bU, ws);
  k_pass3b   <<<8, 256, 0, stream>>>(ws);
  k_pass4_out<<<g2, 256, 0, stream>>>((const f4*)keys, (const f4*)query,
                                      (const f4*)values, aL, aU, bL, bU, ws,
                                      (f4*)out);
}